// DualAttention_5351529251028
// MI455X (gfx1250) — compile-verified
//
#include <hip/hip_runtime.h>
#include <math.h>

typedef __attribute__((ext_vector_type(2))) float v2f;
typedef __attribute__((ext_vector_type(8))) float v8f;

#define BB 4
#define TT 64
#define SS 256
#define DD 512
#define FF 128

// -----------------------------------------------------------------------------
// C[M,N] = tanh( A[M,K] @ W[N,K]^T + bias[N] )   (fp32, V_WMMA_F32_16X16X4_F32)
// One wave32 per 16x16 output tile; 8 waves per block.
// Requires M%16==0, N%16==0, K%4==0 (true for all 4 projections here).
//
// Fragment mapping per ISA 7.12.2 (32-bit A 16x4 / mirrored B 4x16):
//   lane&15 = row (A) / col (B);  lane>>4 selects K pair {k0,k0+1} vs {k0+2,k0+3}
//   -> each lane loads one float2 from A and one from W per k-step.
// -----------------------------------------------------------------------------
__global__ __launch_bounds__(256) void gemm_tanh_wmma(
    const float* __restrict__ A, const float* __restrict__ W,
    const float* __restrict__ bias, float* __restrict__ C,
    int M, int N, int K)
{
  const int lane  = threadIdx.x & 31;
  const int wave  = threadIdx.x >> 5;
  const int tilesN = N >> 4;
  const int w     = blockIdx.x * 8 + wave;
  const int tileM = w / tilesN;
  const int tileN = w % tilesN;
  if (tileM * 16 >= M) return;           // wave-uniform guard

  const int lo = lane & 15;
  const int hi = lane >> 4;              // 0 or 1

  const float* __restrict__ arow = A + (size_t)(tileM * 16 + lo) * K + 2 * hi;
  const float* __restrict__ wrow = W + (size_t)(tileN * 16 + lo) * K + 2 * hi;

  v8f acc = {};
#pragma unroll 4
  for (int k0 = 0; k0 < K; k0 += 4) {
    v2f a = *(const v2f*)(arow + k0);    // A[m, k0+2hi .. k0+2hi+1]
    v2f b = *(const v2f*)(wrow + k0);    // W[n, k0+2hi .. k0+2hi+1]  (== B^T row)
    acc = __builtin_amdgcn_wmma_f32_16x16x4_f32(
        /*neg_a=*/false, a, /*neg_b=*/false, b,
        /*c_mod=*/(short)0, acc, /*reuse_a=*/false, /*reuse_b=*/false);
  }

  // C/D layout: VGPR r -> (M = r + 8*hi, N = lane&15)
  const float bn = bias[tileN * 16 + lo];
  float* __restrict__ crow = C + (size_t)(tileM * 16 + 8 * hi) * N + tileN * 16 + lo;
#pragma unroll
  for (int r = 0; r < 8; ++r)
    crow[(size_t)r * N] = tanhf(acc[r] + bn);
}

// -----------------------------------------------------------------------------
// Fused dual attention. Key identity: alpha*beta / sum_s(alpha*beta) ==
// softmax_s(g1+g2) per channel d (both softmax normalizers cancel). So one
// online-softmax pass over S per (q,d), accumulating sum_s p * enc.
// One thread per (q, d); q = b*T + t in [0, 256), d in [0, 512).
// Writes concat[:, :D] = output, concat[:, D:] = attn, and attn itself.
// -----------------------------------------------------------------------------
__global__ __launch_bounds__(256) void dual_attn_fused(
    const float* __restrict__ out_hs,  // [B*S, D]
    const float* __restrict__ out_fds, // [B*S, D]
    const float* __restrict__ out2,    // [B*T, D]
    const float* __restrict__ out3,    // [B*T, D]
    const float* __restrict__ enc,     // [B*S, D] (raw flat view of [S,B,E])
    const float* __restrict__ outp,    // [B*T, D]
    float* __restrict__ concat,        // [B*T, 2D]
    float* __restrict__ attn_out)      // [B*T, D]
{
  const int gid = blockIdx.x * blockDim.x + threadIdx.x;
  const int d = gid & (DD - 1);
  const int q = gid >> 9;              // log2(DD)
  const int b = q >> 6;                // q / TT

  const float o2 = out2[(size_t)q * DD + d];
  const float o3 = out3[(size_t)q * DD + d];
  const float* __restrict__ hsp = out_hs  + (size_t)b * SS * DD + d;
  const float* __restrict__ fdp = out_fds + (size_t)b * SS * DD + d;
  const float* __restrict__ ep  = enc     + (size_t)b * SS * DD + d;

  float m = -INFINITY, l = 0.f, acc = 0.f;
#pragma unroll 4
  for (int s = 0; s < SS; ++s) {
    const float g  = hsp[(size_t)s * DD] * o2 + fdp[(size_t)s * DD] * o3;
    const float mn = fmaxf(m, g);
    const float sc = __expf(m - mn);   // 0.0 on first iteration (m = -inf); l, acc start at 0
    const float p  = __expf(g - mn);
    l   = l   * sc + p;
    acc = acc * sc + p * ep[(size_t)s * DD];
    m = mn;
  }
  const float attn = acc / l;
  const float ov = outp[(size_t)q * DD + d];
  concat[(size_t)q * 2 * DD + d]       = ov;
  concat[(size_t)q * 2 * DD + DD + d]  = attn;
  attn_out[(size_t)q * DD + d]         = attn;
}

extern "C" void kernel_launch(void* const* d_in, const int* in_sizes, int n_in,
                              void* d_out, int out_size, void* d_ws, size_t ws_size,
                              hipStream_t stream) {
  const float* outp = (const float*)d_in[0];   // [B,T,D]
  const float* enc  = (const float*)d_in[1];   // [S,B,E] == flat [B*S, E]
  const float* inz  = (const float*)d_in[2];   // [B,S,F] == flat [B*S, F]
  const float* W1   = (const float*)d_in[3];
  const float* b1   = (const float*)d_in[4];
  const float* W2   = (const float*)d_in[5];
  const float* b2   = (const float*)d_in[6];
  const float* W3   = (const float*)d_in[7];
  const float* b3   = (const float*)d_in[8];
  const float* W4   = (const float*)d_in[9];
  const float* b4   = (const float*)d_in[10];

  const int M1 = BB * SS;   // 1024
  const int M2 = BB * TT;   // 256

  float* ws      = (float*)d_ws;
  float* out_hs  = ws;                          // 1024*512
  float* out_fds = out_hs  + (size_t)M1 * DD;   // 1024*512
  float* out2    = out_fds + (size_t)M1 * DD;   // 256*512
  float* out3    = out2    + (size_t)M2 * DD;   // 256*512

  float* concat   = (float*)d_out;                        // B*T*2D
  float* attn_out = concat + (size_t)M2 * 2 * DD;         // B*T*D

  const int g1 = (M1 / 16) * (DD / 16) / 8;   // 256 blocks
  const int g2 = (M2 / 16) * (DD / 16) / 8;   // 64 blocks

  gemm_tanh_wmma<<<g1, 256, 0, stream>>>(enc,  W1, b1, out_hs,  M1, DD, DD);
  gemm_tanh_wmma<<<g1, 256, 0, stream>>>(inz,  W3, b3, out_fds, M1, DD, FF);
  gemm_tanh_wmma<<<g2, 256, 0, stream>>>(outp, W2, b2, out2,    M2, DD, DD);
  gemm_tanh_wmma<<<g2, 256, 0, stream>>>(outp, W4, b4, out3,    M2, DD, DD);

  dual_attn_fused<<<(M2 * DD) / 256, 256, 0, stream>>>(
      out_hs, out_fds, out2, out3, enc, outp, concat, attn_out);
}